// GCNII_12343736009311
// MI455X (gfx1250) — compile-verified
//
#include <hip/hip_runtime.h>
#include <math.h>

typedef __attribute__((ext_vector_type(2))) float v2f;
typedef __attribute__((ext_vector_type(8))) float v8f;

#define DIN   128
#define HID   64
#define DOUT  40
#define LAYERS 8
#define ALPHA 0.1f
#define WAVES_PER_WG 8

// D = A(16x4 f32) * B(4x16 f32) + C(16x16 f32), full wave32 op.
__device__ __forceinline__ v8f wmma4(v2f a, v2f b, v8f c) {
  return __builtin_amdgcn_wmma_f32_16x16x4_f32(
      /*neg_a=*/false, a, /*neg_b=*/false, b,
      /*c_mod=*/(short)0, c, /*reuse_a=*/false, /*reuse_b=*/false);
}

// ---------------- normalization setup ----------------

__global__ __launch_bounds__(256) void k_init_deg(float* deg, int n) {
  int i = blockIdx.x * 256 + threadIdx.x;
  if (i < n) deg[i] = 1.0f;  // self-loop
}

__global__ __launch_bounds__(256) void k_count_deg(const int* __restrict__ dst,
                                                   float* deg, int e) {
  int i = blockIdx.x * 256 + threadIdx.x;
  if (i < e) atomicAdd(&deg[dst[i]], 1.0f);
}

__global__ __launch_bounds__(256) void k_rsqrt(float* deg, int n) {
  int i = blockIdx.x * 256 + threadIdx.x;
  if (i < n) deg[i] = rsqrtf(deg[i]);  // deg >= 1 always (self-loops)
}

__global__ __launch_bounds__(256) void k_edgew(const int* __restrict__ src,
                                               const int* __restrict__ dst,
                                               const float* __restrict__ dinv,
                                               float* __restrict__ w, int e) {
  int i = blockIdx.x * 256 + threadIdx.x;
  if (i < e) w[i] = dinv[src[i]] * dinv[dst[i]];
}

// ---------------- propagation (L2-resident scatter) ----------------

// agg[i][:] = dinv[i]^2 * h[i][:]   (self-loop term, also zero-initializes agg)
__global__ __launch_bounds__(256) void k_init_agg(const float* __restrict__ dinv,
                                                  const float* __restrict__ h,
                                                  float* __restrict__ agg, int n) {
  long long t = (long long)blockIdx.x * 256 + threadIdx.x;
  int i = (int)(t >> 4);
  if (i >= n) return;
  int j = ((int)t & 15) << 2;
  float w = dinv[i]; w *= w;
  const float4 hv = *(const float4*)(h + (long long)i * HID + j);
  float4 r; r.x = w * hv.x; r.y = w * hv.y; r.z = w * hv.z; r.w = w * hv.w;
  *(float4*)(agg + (long long)i * HID + j) = r;
}

// 16 threads per edge, float4 each: agg[dst] += w_e * h[src]
__global__ __launch_bounds__(256) void k_scatter(const int* __restrict__ src,
                                                 const int* __restrict__ dst,
                                                 const float* __restrict__ w_e,
                                                 const float* __restrict__ h,
                                                 float* __restrict__ agg, int e) {
  long long t = (long long)blockIdx.x * 256 + threadIdx.x;
  int i = (int)(t >> 4);
  if (i >= e) return;
  int j = ((int)t & 15) << 2;
  int s = src[i], d = dst[i];
  float w = w_e[i];
  const float4 hv = *(const float4*)(h + (long long)s * HID + j);
  float* ap = agg + (long long)d * HID + j;
  atomicAdd(ap + 0, w * hv.x);
  atomicAdd(ap + 1, w * hv.y);
  atomicAdd(ap + 2, w * hv.z);
  atomicAdd(ap + 3, w * hv.w);
}

// ---------------- WMMA GEMMs ----------------
// One wave = one 16-row x 64-col tile; 4x v8f accumulators.
// A layout (16x4 f32): lane<16 -> K={k,k+1}, lane>=16 -> K={k+2,k+3}.
// B layout (4x16 f32): V0 = rows {k, k+2}, V1 = rows {k+1, k+3} across halves.
// W is stored PAIR-INTERLEAVED in LDS: sW[(k>>1)*2*NCOL + c*2 + (k&1)],
// so each B operand {W[ka][c], W[ka+1][c]} is one aligned ds_load_b64.

// h0 = relu(x @ W1 + b1)
__global__ __launch_bounds__(256) void k_gemm_in(const float* __restrict__ x,
                                                 const float* __restrict__ W1,
                                                 const float* __restrict__ b1,
                                                 float* __restrict__ h0, int n) {
  __shared__ float sW[DIN * HID];  // 32 KB, pair-interleaved
  __shared__ float sB[HID];
  for (int i = threadIdx.x; i < DIN * HID; i += 256) {
    int k = i >> 6, c = i & 63;                     // HID == 64
    sW[(k >> 1) * (2 * HID) + c * 2 + (k & 1)] = W1[i];
  }
  if (threadIdx.x < HID) sB[threadIdx.x] = b1[threadIdx.x];
  __syncthreads();

  const int wave = threadIdx.x >> 5, lane = threadIdx.x & 31;
  const int lrow = lane & 15, khalf = lane >> 4;
  const long long rowBase = ((long long)blockIdx.x * WAVES_PER_WG + wave) * 16;
  long long row = rowBase + lrow;
  long long rc = row < n ? row : (long long)(n - 1);
  const float* xr = x + rc * DIN;

  v8f a0 = {}, a1 = {}, a2 = {}, a3 = {};
#pragma unroll
  for (int k = 0; k < DIN; k += 4) {
    const int ka = k + khalf * 2;                   // even
    v2f a; a.x = xr[ka]; a.y = xr[ka + 1];
    const float* bp = sW + (ka >> 1) * (2 * HID) + lrow * 2;
    a0 = wmma4(a, *(const v2f*)(bp),      a0);
    a1 = wmma4(a, *(const v2f*)(bp + 32), a1);
    a2 = wmma4(a, *(const v2f*)(bp + 64), a2);
    a3 = wmma4(a, *(const v2f*)(bp + 96), a3);
  }
  // C layout: VGPR r -> rows {r, r+8} split across lane halves, col = lane&15
  for (int r = 0; r < 8; ++r) {
    long long orow = rowBase + r + khalf * 8;
    if (orow >= n) continue;
    float* op = h0 + orow * HID;
    int c = lrow;
    op[c]      = fmaxf(a0[r] + sB[c],      0.0f);
    op[c + 16] = fmaxf(a1[r] + sB[c + 16], 0.0f);
    op[c + 32] = fmaxf(a2[r] + sB[c + 32], 0.0f);
    op[c + 48] = fmaxf(a3[r] + sB[c + 48], 0.0f);
  }
}

// fused: out = 0.9*agg + 0.1*h0 (computed on the fly);
// hout = relu((1-beta)*out + beta*(out @ W))
__global__ __launch_bounds__(256) void k_gemm_layer(const float* __restrict__ agg,
                                                    const float* __restrict__ h0,
                                                    const float* __restrict__ W,
                                                    float* __restrict__ hout,
                                                    int n, float beta) {
  __shared__ float sW[HID * HID];  // 16 KB, pair-interleaved
  for (int i = threadIdx.x; i < HID * HID; i += 256) {
    int k = i >> 6, c = i & 63;
    sW[(k >> 1) * (2 * HID) + c * 2 + (k & 1)] = W[i];
  }
  __syncthreads();

  const int wave = threadIdx.x >> 5, lane = threadIdx.x & 31;
  const int lrow = lane & 15, khalf = lane >> 4;
  const long long rowBase = ((long long)blockIdx.x * WAVES_PER_WG + wave) * 16;
  long long row = rowBase + lrow;
  long long rc = row < n ? row : (long long)(n - 1);
  const float* ar = agg + rc * HID;
  const float* hr = h0 + rc * HID;
  const float na = 1.0f - ALPHA;

  v8f a0 = {}, a1 = {}, a2 = {}, a3 = {};
#pragma unroll
  for (int k = 0; k < HID; k += 4) {
    const int ka = k + khalf * 2;
    v2f a;
    a.x = na * ar[ka]     + ALPHA * hr[ka];
    a.y = na * ar[ka + 1] + ALPHA * hr[ka + 1];
    const float* bp = sW + (ka >> 1) * (2 * HID) + lrow * 2;
    a0 = wmma4(a, *(const v2f*)(bp),      a0);
    a1 = wmma4(a, *(const v2f*)(bp + 32), a1);
    a2 = wmma4(a, *(const v2f*)(bp + 64), a2);
    a3 = wmma4(a, *(const v2f*)(bp + 96), a3);
  }
  const float nb = 1.0f - beta;
  for (int r = 0; r < 8; ++r) {
    long long orow = rowBase + r + khalf * 8;
    if (orow >= n) continue;
    const float* aR = agg + orow * HID;
    const float* hR = h0 + orow * HID;
    float* op = hout + orow * HID;
    int c = lrow;
    float o;
    o = na * aR[c]      + ALPHA * hR[c];      op[c]      = fmaxf(nb * o + beta * a0[r], 0.0f);
    o = na * aR[c + 16] + ALPHA * hR[c + 16]; op[c + 16] = fmaxf(nb * o + beta * a1[r], 0.0f);
    o = na * aR[c + 32] + ALPHA * hR[c + 32]; op[c + 32] = fmaxf(nb * o + beta * a2[r], 0.0f);
    o = na * aR[c + 48] + ALPHA * hR[c + 48]; op[c + 48] = fmaxf(nb * o + beta * a3[r], 0.0f);
  }
}

// out = h @ W2 + b2   (DOUT=40 padded to 48 cols, 3 col-groups)
__global__ __launch_bounds__(256) void k_gemm_out(const float* __restrict__ h,
                                                  const float* __restrict__ W2,
                                                  const float* __restrict__ b2,
                                                  float* __restrict__ out, int n) {
  __shared__ float sW[HID * 48];  // 12 KB, pair-interleaved, zero-padded cols 40..47
  __shared__ float sB[48];
  for (int i = threadIdx.x; i < HID * 48; i += 256) {
    int k = i / 48, c = i % 48;
    sW[(k >> 1) * 96 + c * 2 + (k & 1)] = (c < DOUT) ? W2[k * DOUT + c] : 0.0f;
  }
  if (threadIdx.x < 48) sB[threadIdx.x] = (threadIdx.x < DOUT) ? b2[threadIdx.x] : 0.0f;
  __syncthreads();

  const int wave = threadIdx.x >> 5, lane = threadIdx.x & 31;
  const int lrow = lane & 15, khalf = lane >> 4;
  const long long rowBase = ((long long)blockIdx.x * WAVES_PER_WG + wave) * 16;
  long long row = rowBase + lrow;
  long long rc = row < n ? row : (long long)(n - 1);
  const float* hr = h + rc * HID;

  v8f a0 = {}, a1 = {}, a2 = {};
#pragma unroll
  for (int k = 0; k < HID; k += 4) {
    const int ka = k + khalf * 2;
    v2f a; a.x = hr[ka]; a.y = hr[ka + 1];
    const float* bp = sW + (ka >> 1) * 96 + lrow * 2;
    a0 = wmma4(a, *(const v2f*)(bp),      a0);
    a1 = wmma4(a, *(const v2f*)(bp + 32), a1);
    a2 = wmma4(a, *(const v2f*)(bp + 64), a2);
  }
  for (int r = 0; r < 8; ++r) {
    long long orow = rowBase + r + khalf * 8;
    if (orow >= n) continue;
    float* op = out + orow * DOUT;
    int c = lrow;
    op[c]      = a0[r] + sB[c];
    op[c + 16] = a1[r] + sB[c + 16];
    if (c + 32 < DOUT) op[c + 32] = a2[r] + sB[c + 32];
  }
}

// ---------------- host ----------------

extern "C" void kernel_launch(void* const* d_in, const int* in_sizes, int n_in,
                              void* d_out, int out_size, void* d_ws, size_t ws_size,
                              hipStream_t stream) {
  const float* x     = (const float*)d_in[0];
  const int*   eidx  = (const int*)d_in[1];   // [2][E] int32 (JAX default x32)
  const float* W1    = (const float*)d_in[2];
  const float* b1    = (const float*)d_in[3];
  const float* convw = (const float*)d_in[4]; // [8][64][64]
  const float* W2    = (const float*)d_in[5];
  const float* b2    = (const float*)d_in[6];

  const int n = in_sizes[0] / DIN;       // 100000
  const int e = in_sizes[1] / 2;         // 1600000
  const int* src = eidx;
  const int* dst = eidx + e;

  char* ws = (char*)d_ws;
  size_t off = 0;
  auto alloc = [&](size_t bytes) -> char* {
    off = (off + 255) & ~(size_t)255;
    char* p = ws + off;
    off += bytes;
    return p;
  };
  float* dinv = (float*)alloc((size_t)n * 4);            // deg -> rsqrt in place
  float* w_e  = (float*)alloc((size_t)e * 4);
  float* h0   = (float*)alloc((size_t)n * HID * 4);
  float* X    = (float*)alloc((size_t)n * HID * 4);      // agg / out
  float* Y    = (float*)alloc((size_t)n * HID * 4);      // h

  const int bN  = (n + 255) / 256;
  const int bE  = (e + 255) / 256;
  const int bN16 = (int)(((long long)n * 16 + 255) / 256);
  const int bE16 = (int)(((long long)e * 16 + 255) / 256);
  const int tiles = (n + 15) / 16;
  const int wgs = (tiles + WAVES_PER_WG - 1) / WAVES_PER_WG;

  // normalization
  k_init_deg <<<bN, 256, 0, stream>>>(dinv, n);
  k_count_deg<<<bE, 256, 0, stream>>>(dst, dinv, e);
  k_rsqrt    <<<bN, 256, 0, stream>>>(dinv, n);
  k_edgew    <<<bE, 256, 0, stream>>>(src, dst, dinv, w_e, e);

  // h0 = relu(x @ W1 + b1)
  k_gemm_in<<<wgs, 256, 0, stream>>>(x, W1, b1, h0, n);

  const float* hin = h0;
  for (int i = 0; i < LAYERS; ++i) {
    k_init_agg<<<bN16, 256, 0, stream>>>(dinv, hin, X, n);
    k_scatter <<<bE16, 256, 0, stream>>>(src, dst, w_e, hin, X, e);
    float beta = logf(0.5f / (float)(i + 1) + 1.0f);
    k_gemm_layer<<<wgs, 256, 0, stream>>>(X, h0, convw + (size_t)i * HID * HID, Y, n, beta);
    hin = Y;
  }

  k_gemm_out<<<wgs, 256, 0, stream>>>(Y, W2, b2, (float*)d_out, n);
}